// TopoSparseObjective_87737591923123
// MI455X (gfx1250) — compile-verified
//
#include <hip/hip_runtime.h>
#include <math.h>

namespace {

constexpr int HH = 512;
constexpr int WW = 512;
constexpr int BB = 8;
constexpr int HW = HH * WW;      // 262144 = 2^18
constexpr int NN = BB * HW;      // 2097152
constexpr int N2 = 2 * NN;       // 4194304 (affinity tensors)
constexpr float EPSL = 1e-6f;
constexpr float ALPHA_F = 0.25f;
constexpr float FINF = 3.402823466e38f;

// fused skeleton-step tile geometry
constexpr int TS  = 32;          // interior tile (32x32, 4 px/thread @ 256 threads)
constexpr int E1D = TS + 4;      // img tile with halo 2  -> 36
constexpr int E1S = E1D + 1;     // padded LDS stride     -> 37
constexpr int E2D = TS + 2;      // eroded tile, halo 1   -> 34
constexpr int E2S = E2D + 1;     // padded LDS stride     -> 35

typedef __attribute__((ext_vector_type(2))) float v2f;
typedef __attribute__((ext_vector_type(8))) float v8f;

__device__ __forceinline__ float sigm(float x) { return 1.0f / (1.0f + expf(-x)); }

// stable elementwise focal term given logits x, target t
__device__ __forceinline__ float focal_elem(float x, float t) {
    float bce = fmaxf(x, 0.0f) - x * t + log1pf(expf(-fabsf(x)));
    float p   = sigm(x);
    float pt  = p * t + (1.0f - p) * (1.0f - t);
    float at  = ALPHA_F * t + (1.0f - ALPHA_F) * (1.0f - t);
    float om  = 1.0f - pt;
    return at * om * om * bce;
}

__device__ __forceinline__ float blockReduce256(float v, float* sred) {
    const int tid = threadIdx.x;
    sred[tid] = v;
    __syncthreads();
    for (int off = 128; off > 0; off >>= 1) {
        if (tid < off) sred[tid] += sred[tid + off];
        __syncthreads();
    }
    float r = sred[0];
    __syncthreads();
    return r;
}

// ---------------------------------------------------------------------------
// Stage A: one fused pointwise pass over all N-sized tensors.
// 256 blocks x 256 threads; 32 blocks per batch image (8192 elems/block)
// so per-batch dice sums never straddle batches.
// Also materializes bufA = sigmoid(mask_logits) and zeroes both skeletons.
//  q0 focal(mask)  q1 focal(junction)  q2 focal(endpoint)  q3 unc MSE
//  q4 sum pm*tm    q5 sum pm           q6 sum tm            (dice mask)
//  q7 sum ps*ts    q8 sum ps           q9 sum ts            (dice skel)
// ---------------------------------------------------------------------------
__global__ void stageA_kernel(const float* __restrict__ mlog, const float* __restrict__ mask,
                              const float* __restrict__ slog, const float* __restrict__ skl,
                              const float* __restrict__ jlog, const float* __restrict__ junc,
                              const float* __restrict__ elog, const float* __restrict__ endp,
                              const float* __restrict__ ulog, const float* __restrict__ unc,
                              float* __restrict__ bufA, float* __restrict__ skelP,
                              float* __restrict__ skelT, float* __restrict__ pw) {
    const int blk = blockIdx.x;
    const int tid = threadIdx.x;
    const int per = NN / 256;            // 8192
    const int start = blk * per;

    float q[10];
#pragma unroll
    for (int i = 0; i < 10; ++i) q[i] = 0.0f;

    for (int i = tid; i < per; i += 256) {
        const int idx = start + i;
        float x = mlog[idx], t = mask[idx];
        float p = sigm(x);
        q[0] += focal_elem(x, t);
        q[4] += p * t;
        q[5] += p;
        q[6] += t;
        bufA[idx]  = p;     // skeletonize input (already in [0,1])
        skelP[idx] = 0.0f;
        skelT[idx] = 0.0f;
        float xs = slog[idx], ts = skl[idx];
        float ps = sigm(xs);
        q[7] += ps * ts;
        q[8] += ps;
        q[9] += ts;
        q[1] += focal_elem(jlog[idx], junc[idx]);
        q[2] += focal_elem(elog[idx], endp[idx]);
        float du = sigm(ulog[idx]) - unc[idx];
        q[3] += du * du;
    }

    __shared__ float sred[256];
    for (int qq = 0; qq < 10; ++qq) {
        float r = blockReduce256(q[qq], sred);
        if (tid == 0) pw[blk * 10 + qq] = r;
    }
}

// dst = clip(src, 0, 1) -- init for target skeletonize
__global__ void clipInit_kernel(const float* __restrict__ src, float* __restrict__ dst) {
    int idx = blockIdx.x * 256 + threadIdx.x;
    if (idx >= NN) return;
    dst[idx] = fminf(fmaxf(src[idx], 0.0f), 1.0f);
}

// ---------------------------------------------------------------------------
// Fused skeletonize step (one per chain link a_i -> a_{i+1}):
//   LDS: sE1 = img tile (halo 2, OOB=+inf -> min-pool window shrink)
//        sE2 = erode(sE1) tile (halo 1, outside-image=-inf -> max-pool pad)
//   per interior pixel: imgOut = erode center (a_{i+1});
//   delta = relu(a_i - maxpool3(a_{i+1})); skel += relu(delta - skel*delta)
// 2048 blocks (16x16 tiles x 8 batches) x 256 threads.
// ---------------------------------------------------------------------------
__global__ void skelStep_kernel(const float* __restrict__ img, float* __restrict__ imgOut,
                                float* __restrict__ skel) {
    __shared__ float sE1[E1D * E1S];
    __shared__ float sE2[E2D * E2S];
    const int tid = threadIdx.x;
    const int blk = blockIdx.x;          // 0..2047
    const int tx  = blk & 15;
    const int ty  = (blk >> 4) & 15;
    const int b   = blk >> 8;
    const int gx0 = tx * TS;
    const int gy0 = ty * TS;
    const int base = b << 18;            // b * HW

    // phase 1: stage img tile (halo 2); out-of-image -> +inf
    for (int i = tid; i < E1D * E1D; i += 256) {
        int ly = i / E1D, lx = i - ly * E1D;
        int gy = gy0 - 2 + ly;
        int gx = gx0 - 2 + lx;
        float v = FINF;
        if (gy >= 0 && gy < HH && gx >= 0 && gx < WW)
            v = img[base + (gy << 9) + gx];
        sE1[ly * E1S + lx] = v;
    }
    __syncthreads();

    // phase 2: 5-point cross min into sE2 (halo 1); out-of-image -> -inf
    for (int i = tid; i < E2D * E2D; i += 256) {
        int ly = i / E2D, lx = i - ly * E2D;
        int gy = gy0 - 1 + ly;
        int gx = gx0 - 1 + lx;
        float v = -FINF;
        if (gy >= 0 && gy < HH && gx >= 0 && gx < WW) {
            const float* c = &sE1[(ly + 1) * E1S + (lx + 1)];
            v = c[0];
            v = fminf(v, c[-E1S]);
            v = fminf(v, c[ E1S]);
            v = fminf(v, c[-1]);
            v = fminf(v, c[ 1]);
        }
        sE2[ly * E2S + lx] = v;
    }
    __syncthreads();

    // phase 3: interior: emit a_{i+1}, 3x3 max, skeleton update
    for (int i = tid; i < TS * TS; i += 256) {
        int ly = i >> 5, lx = i & 31;
        int gy = gy0 + ly, gx = gx0 + lx;
        int gidx = base + (gy << 9) + gx;
        const float* c2 = &sE2[(ly + 1) * E2S + (lx + 1)];
        float e2c = c2[0];
        imgOut[gidx] = e2c;
        float o = e2c;
        o = fmaxf(o, c2[-1]);        o = fmaxf(o, c2[1]);
        o = fmaxf(o, c2[-E2S - 1]);  o = fmaxf(o, c2[-E2S]);  o = fmaxf(o, c2[-E2S + 1]);
        o = fmaxf(o, c2[ E2S - 1]);  o = fmaxf(o, c2[ E2S]);  o = fmaxf(o, c2[ E2S + 1]);
        float e1c = sE1[(ly + 2) * E1S + (lx + 2)];
        float delta = fmaxf(e1c - o, 0.0f);
        float s = skel[gidx];
        skel[gidx] = s + fmaxf(delta - s * delta, 0.0f);
    }
}

// clDice per-batch sums: q0=sum(ps*mask) q1=sum(ps) q2=sum(ts*pred_probs) q3=sum(ts)
__global__ void clSums_kernel(const float* __restrict__ skelP, const float* __restrict__ skelT,
                              const float* __restrict__ mask, const float* __restrict__ mlog,
                              float* __restrict__ clp) {
    const int blk = blockIdx.x;
    const int tid = threadIdx.x;
    const int per = NN / 256;
    const int start = blk * per;
    float q0 = 0.f, q1 = 0.f, q2 = 0.f, q3 = 0.f;
    for (int i = tid; i < per; i += 256) {
        int idx = start + i;
        float ps = fminf(fmaxf(skelP[idx], 0.0f), 1.0f);
        float ts = fminf(fmaxf(skelT[idx], 0.0f), 1.0f);
        q0 += ps * mask[idx];
        q1 += ps;
        q2 += ts * sigm(mlog[idx]);
        q3 += ts;
    }
    __shared__ float sred[256];
    float r;
    r = blockReduce256(q0, sred); if (tid == 0) clp[blk * 4 + 0] = r;
    r = blockReduce256(q1, sred); if (tid == 0) clp[blk * 4 + 1] = r;
    r = blockReduce256(q2, sred); if (tid == 0) clp[blk * 4 + 2] = r;
    r = blockReduce256(q3, sred); if (tid == 0) clp[blk * 4 + 3] = r;
}

// masked smooth-L1 partial sums over the (8,2,512,512) affinity tensors
__global__ void affinity_kernel(const float* __restrict__ ap, const float* __restrict__ at,
                                const float* __restrict__ skl, float* __restrict__ affp) {
    const int blk = blockIdx.x;
    const int tid = threadIdx.x;
    const int per = N2 / 256;            // 16384
    const int start = blk * per;
    float acc = 0.0f;
    for (int i = tid; i < per; i += 256) {
        int e   = start + i;
        int pix = e & (HW - 1);
        int b   = e >> 19;               // e / (2*HW)
        float m = skl[(b << 18) + pix];  // broadcast over channel dim
        float d  = ap[e] * m - at[e] * m;
        float ad = fabsf(d);
        acc += (ad < 1.0f) ? 0.5f * d * d : ad - 0.5f;
    }
    __shared__ float sred[256];
    float r = blockReduce256(acc, sred);
    if (tid == 0) affp[blk] = r;
}

// ---------------------------------------------------------------------------
// Final: one wave of 32 threads. Global 256-entry partial sums reduced
// losslessly with V_WMMA_F32_16X16X4_F32 (A = 16x4 f32 chunk of partials,
// B = ones => D rows are exact f32 row sums, accumulated through C).
// Per-batch dice/clDice sums reduced in fixed order on lanes 0-7.
// ---------------------------------------------------------------------------
__global__ void final_kernel(const float* __restrict__ pw, const float* __restrict__ clp,
                             const float* __restrict__ affp, float* __restrict__ out) {
    const int lane = threadIdx.x;   // 0..31, full wave, no divergence before WMMA
    const int m  = lane & 15;
    const int kb = (lane >> 4) << 1;

    v2f bones = {1.0f, 1.0f};
    float totals[5];                // 0:focal_mask 1:focal_junc 2:focal_end 3:unc 4:affinity

    for (int q = 0; q < 5; ++q) {   // uniform loop: EXEC stays all-ones
        v8f c = {0.f, 0.f, 0.f, 0.f, 0.f, 0.f, 0.f, 0.f};
        for (int j = 0; j < 4; ++j) {
            int i0 = j * 64 + m * 4 + kb;
            v2f a;
            if (q < 4) {
                a.x = pw[i0 * 10 + q];
                a.y = pw[(i0 + 1) * 10 + q];
            } else {
                a.x = affp[i0];
                a.y = affp[i0 + 1];
            }
            c = __builtin_amdgcn_wmma_f32_16x16x4_f32(
                    /*neg_a=*/false, a, /*neg_b=*/false, bones,
                    /*c_mod=*/(short)0, c, /*reuse_a=*/false, /*reuse_b=*/false);
        }
        float s = c[0] + c[1] + c[2] + c[3] + c[4] + c[5] + c[6] + c[7];
        totals[q] = __shfl(s, 0, 32) + __shfl(s, 16, 32);
    }

    // per-batch sums: lanes 0..7 each own one batch (32 blocks, fixed order)
    float pb[10];
#pragma unroll
    for (int i = 0; i < 10; ++i) pb[i] = 0.0f;
    if (lane < 8) {
        for (int k = 0; k < 32; ++k) {
            int blk = lane * 32 + k;
#pragma unroll
            for (int qq = 0; qq < 6; ++qq) pb[qq] += pw[blk * 10 + 4 + qq];
#pragma unroll
            for (int qq = 0; qq < 4; ++qq) pb[6 + qq] += clp[blk * 4 + qq];
        }
    }
    float diceM = (2.0f * pb[0] + EPSL) / (pb[1] + pb[2] + EPSL);
    float diceS = (2.0f * pb[3] + EPSL) / (pb[4] + pb[5] + EPSL);
    float prec  = pb[6] / (pb[7] + EPSL);
    float sens  = pb[8] / (pb[9] + EPSL);
    float clv   = (2.0f * prec * sens + EPSL) / (prec + sens + EPSL);
    float tskel = pb[5];                // per-batch sum of skeleton input

    float dM = 0.f, dS = 0.f, cs = 0.f, tk = 0.f;
    for (int b = 0; b < 8; ++b) {
        dM += __shfl(diceM, b, 32);
        dS += __shfl(diceS, b, 32);
        cs += __shfl(clv,   b, 32);
        tk += __shfl(tskel, b, 32);
    }

    const float invN = 1.0f / (float)NN;
    float mask_loss     = (1.0f - dM / 8.0f) + totals[0] * invN;
    float skeleton_loss = 1.0f - dS / 8.0f;
    float topo_loss     = 1.0f - cs / 8.0f;
    float node_loss     = 0.5f * (totals[1] * invN + totals[2] * invN);
    float msum          = 2.0f * tk;
    float aff_loss      = (msum == 0.0f) ? 0.0f : totals[4] / fmaxf(msum, 1.0f);
    float unc_loss      = totals[3] * invN;

    float total = mask_loss + skeleton_loss + 0.5f * topo_loss +
                  0.5f * node_loss + 0.5f * aff_loss + 0.1f * unc_loss;
    if (lane == 0) out[0] = total;
}

} // anonymous namespace

extern "C" void kernel_launch(void* const* d_in, const int* in_sizes, int n_in,
                              void* d_out, int out_size, void* d_ws, size_t ws_size,
                              hipStream_t stream) {
    (void)in_sizes; (void)n_in; (void)out_size; (void)ws_size;

    const float* mlog  = (const float*)d_in[0];
    const float* slog  = (const float*)d_in[1];
    const float* ulog  = (const float*)d_in[2];
    const float* jlog  = (const float*)d_in[3];
    const float* elog  = (const float*)d_in[4];
    const float* apred = (const float*)d_in[5];
    const float* mask  = (const float*)d_in[6];
    const float* skl   = (const float*)d_in[7];
    const float* junc  = (const float*)d_in[8];
    const float* endp  = (const float*)d_in[9];
    const float* atarg = (const float*)d_in[10];
    const float* unc   = (const float*)d_in[11];

    float* ws    = (float*)d_ws;
    float* bufA  = ws;                       // img chain ping
    float* bufB  = ws + (size_t)NN;          // img chain pong
    float* skelP = ws + 2 * (size_t)NN;      // skeleton of pred probs
    float* skelT = ws + 3 * (size_t)NN;      // skeleton of mask
    float* pw    = ws + 4 * (size_t)NN;      // 256 x 10 partials
    float* clp   = pw + 256 * 10;            // 256 x 4 partials
    float* affp  = clp + 256 * 4;            // 256 partials

    const int NB  = NN / 256;                // elementwise grid
    const int NBS = (HH / TS) * (WW / TS) * BB;   // 16*16*8 = 2048 tiles

    // Fused pointwise losses + dice partials + prob materialization + skel zeroing
    stageA_kernel<<<256, 256, 0, stream>>>(mlog, mask, slog, skl, jlog, junc,
                                           elog, endp, ulog, unc,
                                           bufA, skelP, skelT, pw);
    // Affinity smooth-L1 partials (independent of skeletonize)
    affinity_kernel<<<256, 256, 0, stream>>>(apred, atarg, skl, affp);

    // --- soft_skeletonize(pred_probs) -> skelP : 11 fused LDS-tiled steps ---
    {
        float* cur = bufA; float* nxt = bufB;
        for (int st = 0; st < 11; ++st) {
            skelStep_kernel<<<NBS, 256, 0, stream>>>(cur, nxt, skelP);
            float* t = cur; cur = nxt; nxt = t;
        }
    }

    // --- soft_skeletonize(clip(mask)) -> skelT ---
    {
        clipInit_kernel<<<NB, 256, 0, stream>>>(mask, bufA);
        float* cur = bufA; float* nxt = bufB;
        for (int st = 0; st < 11; ++st) {
            skelStep_kernel<<<NBS, 256, 0, stream>>>(cur, nxt, skelT);
            float* t = cur; cur = nxt; nxt = t;
        }
    }

    // clDice per-batch partial sums
    clSums_kernel<<<256, 256, 0, stream>>>(skelP, skelT, mask, mlog, clp);

    // Deterministic final reduction + scalar math (WMMA f32 reducer), one wave
    final_kernel<<<1, 32, 0, stream>>>(pw, clp, affp, (float*)d_out);
}